// VolumeRenderer_65180423684381
// MI455X (gfx1250) — compile-verified
//
#include <hip/hip_runtime.h>

// NeRF-style coarse+fine volume renderer for MI455X (gfx1250, wave32).
// Bandwidth-bound (~736 MB @ 23.3 TB/s => ~32us floor). One wave per ray,
// 16 rays (16 waves / 512 threads) per block. All prefix scans are done as
// triangular-matrix multiplies on the f32 WMMA unit (V_WMMA_F32_16X16X4_F32):
// exclusive row-scan of the 16x32 lane-partial matrix = S x U_strict.

#define NRAYS 65536
#define NC 256
#define NI 128
#define NF 384
#define FARV 1e10f

typedef __attribute__((ext_vector_type(2))) float v2f;
typedef __attribute__((ext_vector_type(8))) float v8f;

struct Smem {
  float Zs[16 * 256];    // coarse z per ray (alive until merge)
  float R[16 * 384];     // phase 1: CDF (stride 256); phase 2: z_fine (stride 384)
  float ZSmp[16 * 128];  // importance samples
  float Lsum[16 * 32];   // per-(ray,lane) partial sums (scan input)
  float Loff[16 * 32];   // per-(ray,lane) exclusive offsets (scan output)
};

// Exclusive scan along each row of the 16x32 matrix Lsum -> Loff, computed by
// wave 0 as chained V_WMMA_F32_16X16X4_F32:
//   E1 = M1 * Ustrict                 (cols 0..15)
//   E2 = M1 * Jones + M2 * Ustrict    (cols 16..31)
// A layout (16x4 f32): lane l holds row M=l&15, vgpr j holds K = j + 2*(l>>4).
// B layout (4x16 f32): lane l holds col N=l&15, vgpr j holds K = j + 2*(l>>4).
// C/D layout: lane l holds col N=l&15, vgpr v holds row M = v + 8*(l>>4).
__device__ __forceinline__ void wmma_row_scan_16x32(Smem& sm, int tid) {
  __syncthreads();
  if (tid < 32) {                 // whole wave 0 -> EXEC all ones (WMMA requirement)
    const int lane = tid;
    const int M  = lane & 15;
    const int N  = lane & 15;
    const int hi = lane >> 4;
    const int k0 = 2 * hi;
    v8f e1 = {};
    v8f e2 = {};
#pragma unroll
    for (int kk = 0; kk < 16; kk += 4) {
      v2f a;  a.x  = sm.Lsum[M * 32 + kk + k0];       a.y  = sm.Lsum[M * 32 + kk + k0 + 1];
      v2f a2; a2.x = sm.Lsum[M * 32 + 16 + kk + k0];  a2.y = sm.Lsum[M * 32 + 16 + kk + k0 + 1];
      v2f bu; bu.x = ((kk + k0) < N) ? 1.0f : 0.0f;   bu.y = ((kk + k0 + 1) < N) ? 1.0f : 0.0f;
      v2f bo; bo.x = 1.0f; bo.y = 1.0f;
      e1 = __builtin_amdgcn_wmma_f32_16x16x4_f32(false, a,  false, bu, (short)0, e1, false, false);
      e2 = __builtin_amdgcn_wmma_f32_16x16x4_f32(false, a,  false, bo, (short)0, e2, false, false);
      e2 = __builtin_amdgcn_wmma_f32_16x16x4_f32(false, a2, false, bu, (short)0, e2, false, false);
    }
#pragma unroll
    for (int v = 0; v < 8; ++v) {
      int row = v + 8 * hi;
      sm.Loff[row * 32 + N]      = e1[v];
      sm.Loff[row * 32 + 16 + N] = e2[v];
    }
  }
  __syncthreads();
}

__global__ __launch_bounds__(512) void VolumeRenderer_kernel(
    const float* __restrict__ z_vals, const float* __restrict__ density,
    const float* __restrict__ feature, const float* __restrict__ fine_density,
    const float* __restrict__ fine_feature, const float* __restrict__ u_rand,
    float* __restrict__ out) {
  __shared__ Smem sm;
  const int tid  = threadIdx.x;
  const int wi   = tid >> 5;          // wave (= ray within block)
  const int lane = tid & 31;
  const int r    = blockIdx.x * 16 + wi;
  const int c0   = lane * 8;          // coarse chunk start
  const int f0   = lane * 12;         // fine chunk start

  // Warm L2 for the fine-pass streams while we do the coarse pass.
  __builtin_prefetch(fine_density + (size_t)r * NF + f0, 0, 0);
  __builtin_prefetch(fine_feature + ((size_t)r * NF + f0) * 3, 0, 0);
  __builtin_prefetch(fine_feature + ((size_t)r * NF + f0) * 3 + 16, 0, 0);

  // ---------------- coarse pass: scan + weighted reduction ----------------
  union { float4 v[2]; float f[8]; } zu, du;
  {
    const float4* zp = reinterpret_cast<const float4*>(z_vals + (size_t)r * NC + c0);
    const float4* dp = reinterpret_cast<const float4*>(density + (size_t)r * NC + c0);
    zu.v[0] = zp[0]; zu.v[1] = zp[1];
    du.v[0] = dp[0]; du.v[1] = dp[1];
  }
  float znext = __shfl_down(zu.f[0], 1);
  float s[8], pre[8];
  float run = 0.0f;
#pragma unroll
  for (int j = 0; j < 8; ++j) {
    float dz = (j < 7) ? (zu.f[j + 1] - zu.f[j])
                       : ((lane < 31) ? (znext - zu.f[7]) : FARV);
    s[j] = du.f[j] * dz;
    run += s[j];
    pre[j] = run;
  }
  sm.Lsum[wi * 32 + lane] = run;
#pragma unroll
  for (int j = 0; j < 8; ++j) sm.Zs[wi * 256 + c0 + j] = zu.f[j];

  wmma_row_scan_16x32(sm, tid);                 // WMMA scan #1
  const float off1 = sm.Loff[wi * 32 + lane];

  union { float4 v[6]; float f[24]; } fu;
  {
    const float4* fp = reinterpret_cast<const float4*>(feature + ((size_t)r * NC + c0) * 3);
#pragma unroll
    for (int q = 0; q < 6; ++q) fu.v[q] = fp[q];
  }

  float w[8];
  float fa = 0.f, fb = 0.f, fc = 0.f, fd = 0.f;
#pragma unroll
  for (int j = 0; j < 8; ++j) {
    float Pex = off1 + pre[j] - s[j];           // exclusive cumsum of sigma*delta
    float T   = __expf(-Pex);                   // transmittance
    float ep  = __expf(-s[j]);
    w[j] = T * (1.0f - ep) + 1e-10f;
    fa += w[j] * fu.f[3 * j + 0];
    fb += w[j] * fu.f[3 * j + 1];
    fc += w[j] * fu.f[3 * j + 2];
    fd += w[j] * zu.f[j];
  }
#pragma unroll
  for (int o = 16; o > 0; o >>= 1) {
    fa += __shfl_down(fa, o); fb += __shfl_down(fb, o);
    fc += __shfl_down(fc, o); fd += __shfl_down(fd, o);
  }
  if (lane == 0)
    *reinterpret_cast<float4*>(out + (size_t)r * 8) = make_float4(fa, fb, fc, fd);

  // ---------------- CDF over interior weights w[1..254] ----------------
  float lp[8];
  run = 0.0f;
#pragma unroll
  for (int j = 0; j < 8; ++j) {
    int idx = c0 + j;
    float gv = (idx >= 1 && idx <= 254) ? (w[j] + 1e-5f) : 0.0f;
    run += gv;
    lp[j] = run;
  }
  sm.Lsum[wi * 32 + lane] = run;
  wmma_row_scan_16x32(sm, tid);                 // WMMA scan #2
  const float off2 = sm.Loff[wi * 32 + lane];
  const float Wtot = __shfl(off2, 31) + __shfl(run, 31);
  const float invW = 1.0f / Wtot;
  float* CDF = sm.R + wi * 256;                 // CDF[idx], idx = 0..254 used
#pragma unroll
  for (int j = 0; j < 8; ++j) CDF[c0 + j] = (off2 + lp[j]) * invW;
  __syncthreads();

  // ---------------- importance sampling (searchsorted right + lerp) --------
  const float* Zr = sm.Zs + wi * 256;
  float* zs = sm.ZSmp + wi * 128;
  {
    const float4 uu = *reinterpret_cast<const float4*>(u_rand + (size_t)r * NI + lane * 4);
    const float u4[4] = {uu.x, uu.y, uu.z, uu.w};
#pragma unroll
    for (int m = 0; m < 4; ++m) {
      float u = u4[m];
      int lo = 0, hiq = 255;                    // cdf length 255
      while (lo < hiq) { int mid = (lo + hiq) >> 1; if (CDF[mid] <= u) lo = mid + 1; else hiq = mid; }
      int below = lo - 1;                       // lo >= 1 since CDF[0]==0 <= u
      int above = (lo > 254) ? 254 : lo;
      float cb = CDF[below], ca = CDF[above];
      float den = ca - cb; if (den < 1e-5f) den = 1.0f;
      float t  = (u - cb) / den;
      float zb = 0.5f * (Zr[below] + Zr[below + 1]);   // z_mid bins
      float za = 0.5f * (Zr[above] + Zr[above + 1]);
      zs[lane * 4 + m] = zb + t * (za - zb);
    }
  }
  __syncthreads();                              // CDF reads done; R reusable

  // ---------------- bitonic sort of 128 importance samples (LDS) ----------
  for (int k = 2; k <= 128; k <<= 1) {
    for (int j2 = k >> 1; j2 > 0; j2 >>= 1) {
#pragma unroll
      for (int m = 0; m < 4; ++m) {
        int i = m * 32 + lane;
        int ixj = i ^ j2;
        if (ixj > i) {
          bool up = ((i & k) == 0);
          float a = zs[i], b = zs[ixj];
          if ((a > b) == up) { zs[i] = b; zs[ixj] = a; }
        }
      }
      __syncthreads();
    }
  }

  // ---------------- stable rank-merge: z_fine = sort([z_vals, z_samples]) --
  float* ZF = sm.R + wi * 384;                  // overlays dead CDF region
#pragma unroll
  for (int j = 0; j < 8; ++j) {
    float v = Zr[c0 + j];
    int lo = 0, hiq = 128;                      // count zs strictly < v
    while (lo < hiq) { int mid = (lo + hiq) >> 1; if (zs[mid] < v) lo = mid + 1; else hiq = mid; }
    ZF[c0 + j + lo] = v;
  }
#pragma unroll
  for (int m = 0; m < 4; ++m) {
    int q = lane * 4 + m;
    float v = zs[q];
    int lo = 0, hiq = 256;                      // count z_vals <= v
    while (lo < hiq) { int mid = (lo + hiq) >> 1; if (Zr[mid] <= v) lo = mid + 1; else hiq = mid; }
    ZF[q + lo] = v;
  }
  __syncthreads();

  // ---------------- fine pass: scan + weighted reduction over 384 ----------
  float zf[13];
#pragma unroll
  for (int j = 0; j < 12; ++j) zf[j] = ZF[f0 + j];
  zf[12] = (lane < 31) ? ZF[f0 + 12] : 0.0f;

  union { float4 v[3]; float f[12]; } fdv;
  {
    const float4* fdp = reinterpret_cast<const float4*>(fine_density + (size_t)r * NF + f0);
    fdv.v[0] = fdp[0]; fdv.v[1] = fdp[1]; fdv.v[2] = fdp[2];
  }
  float s2[12], pr2[12];
  run = 0.0f;
#pragma unroll
  for (int j = 0; j < 12; ++j) {
    float dz = (j < 11) ? (zf[j + 1] - zf[j])
                        : ((lane < 31) ? (zf[12] - zf[11]) : FARV);
    s2[j] = fdv.f[j] * dz;
    run += s2[j];
    pr2[j] = run;
  }
  sm.Lsum[wi * 32 + lane] = run;
  wmma_row_scan_16x32(sm, tid);                 // WMMA scan #3
  const float off3 = sm.Loff[wi * 32 + lane];

  union { float4 v[9]; float f[36]; } ffu;
  {
    const float4* ffp = reinterpret_cast<const float4*>(fine_feature + ((size_t)r * NF + f0) * 3);
#pragma unroll
    for (int q = 0; q < 9; ++q) ffu.v[q] = ffp[q];
  }

  fa = fb = fc = fd = 0.0f;
#pragma unroll
  for (int j = 0; j < 12; ++j) {
    float Pex = off3 + pr2[j] - s2[j];
    float T   = __expf(-Pex);
    float ep  = __expf(-s2[j]);
    float wv  = T * (1.0f - ep) + 1e-10f;
    fa += wv * ffu.f[3 * j + 0];
    fb += wv * ffu.f[3 * j + 1];
    fc += wv * ffu.f[3 * j + 2];
    fd += wv * zf[j];
  }
#pragma unroll
  for (int o = 16; o > 0; o >>= 1) {
    fa += __shfl_down(fa, o); fb += __shfl_down(fb, o);
    fc += __shfl_down(fc, o); fd += __shfl_down(fd, o);
  }
  if (lane == 0)
    *reinterpret_cast<float4*>(out + (size_t)r * 8 + 4) = make_float4(fa, fb, fc, fd);
}

extern "C" void kernel_launch(void* const* d_in, const int* in_sizes, int n_in,
                              void* d_out, int out_size, void* d_ws, size_t ws_size,
                              hipStream_t stream) {
  const float* z_vals       = (const float*)d_in[0];
  const float* density      = (const float*)d_in[1];
  const float* feature      = (const float*)d_in[2];
  const float* fine_density = (const float*)d_in[3];
  const float* fine_feature = (const float*)d_in[4];
  const float* u_rand       = (const float*)d_in[5];
  float* out = (float*)d_out;
  (void)in_sizes; (void)n_in; (void)out_size; (void)d_ws; (void)ws_size;

  dim3 grid(NRAYS / 16);   // 16 rays per block (16 waves x 32 lanes)
  dim3 block(512);
  hipLaunchKernelGGL(VolumeRenderer_kernel, grid, block, 0, stream,
                     z_vals, density, feature, fine_density, fine_feature, u_rand, out);
}